// RoIPool_42133629173786
// MI455X (gfx1250) — compile-verified
//
#include <hip/hip_runtime.h>
#include <math.h>

// RoIPool forward, MI455X (gfx1250, wave32).
// Memory-latency-bound gather/max — no matmul, so no WMMA path applies.
// Design: flat thread == one output element in (K, C, 7, 7) order
//   -> output stores fully coalesced (12.8 MB streaming)
//   -> lanes within a wave share a channel's feature rows (overlapping
//      windows) -> L0/L2 friendly; whole 10 MB input is L2-resident.
// global_prefetch_b8 (gfx1250) pulls the next window row during reduction.

#define POOL_H 7
#define POOL_W 7
#define CHANS  256
#define FEAT_H 50
#define FEAT_W 50
#define SP_SCALE 0.0625f

__global__ __launch_bounds__(256) void roipool_fwd_kernel(
    const float* __restrict__ input,      // (B, C, H, W)
    const float* __restrict__ proposals,  // (K, 5) = {b, x1, y1, x2, y2}
    float* __restrict__ out,              // (K, C, 7, 7)
    int total)
{
    int o = blockIdx.x * blockDim.x + threadIdx.x;
    if (o >= total) return;

    // Decode (k, c, ph, pw) from flat index, (K,C,7,7) row-major.
    int pw = o % POOL_W;
    int t  = o / POOL_W;
    int ph = t % POOL_H;
    t /= POOL_H;
    int c  = t % CHANS;
    int k  = t / CHANS;

    // Proposal decode — mostly wave-uniform (49*256 outputs per ROI),
    // so these 5 loads hit L0 for all but one lane group.
    const float* p = proposals + (size_t)k * 5;
    int b  = (int)p[0];
    // jnp.round == round-half-even -> rintf -> v_rndne_f32
    int x1 = (int)rintf(p[1] * SP_SCALE);
    int y1 = (int)rintf(p[2] * SP_SCALE);
    int x2 = (int)rintf(p[3] * SP_SCALE);
    int y2 = (int)rintf(p[4] * SP_SCALE);

    float roi_w = (float)max(x2 - x1 + 1, 1);
    float roi_h = (float)max(y2 - y1 + 1, 1);
    // Keep true IEEE division (bin edges depend on exact rounding).
    float bin_h = roi_h / (float)POOL_H;
    float bin_w = roi_w / (float)POOL_W;

    int hs = min(max((int)floorf((float)ph * bin_h) + y1, 0), FEAT_H);
    int he = min(max((int)ceilf((float)(ph + 1) * bin_h) + y1, 0), FEAT_H);
    int ws = min(max((int)floorf((float)pw * bin_w) + x1, 0), FEAT_W);
    int we = min(max((int)ceilf((float)(pw + 1) * bin_w) + x1, 0), FEAT_W);

    const float* feat = input + (((size_t)b * CHANS + c) * FEAT_H) * FEAT_W;

    bool nonempty = (he > hs) && (we > ws);
    float m = -INFINITY;
    for (int h = hs; h < he; ++h) {
        const float* row = feat + h * FEAT_W + ws;
        // Prefetch next row's window while reducing this one
        // (gfx1250 global_prefetch_b8; speculative, OOB-safe).
        __builtin_prefetch(row + FEAT_W, 0, 0);
        for (int w = 0; w < we - ws; ++w) {
            m = fmaxf(m, row[w]);
        }
    }
    out[o] = nonempty ? m : 0.0f;
}

extern "C" void kernel_launch(void* const* d_in, const int* in_sizes, int n_in,
                              void* d_out, int out_size, void* d_ws, size_t ws_size,
                              hipStream_t stream) {
    const float* input     = (const float*)d_in[0];
    const float* proposals = (const float*)d_in[1];
    float*       out       = (float*)d_out;

    // K from proposals size (K,5); total outputs = K * C * 7 * 7.
    int K = in_sizes[1] / 5;
    int total = K * CHANS * POOL_H * POOL_W;

    const int block = 256;  // 8 wave32 waves
    int grid = (total + block - 1) / block;
    roipool_fwd_kernel<<<grid, block, 0, stream>>>(input, proposals, out, total);
}